// TropicalAttention_23888608101015
// MI455X (gfx1250) — compile-verified
//
#include <hip/hip_runtime.h>
#include <hip/hip_bf16.h>
#include <stdint.h>

// Problem constants (from reference setup_inputs)
#define BATCH 2
#define SEQ   384
#define DMOD  512
#define NH    8
#define DK    64
#define DKP   (DK + 4)       // padded LDS row stride: 16B-aligned rows, banks (4r+d)%64
#define BH    (BATCH * NH)   // 16
#define TI    16
#define TJ    16

typedef __attribute__((ext_vector_type(2))) float v2f;
typedef __attribute__((ext_vector_type(8))) float v8f;

// ---------------------------------------------------------------------------
// Kernel 1: t = log1p(relu(x)), permuted [B,S,D] -> [BH, S, dk]
// ---------------------------------------------------------------------------
__global__ __launch_bounds__(256) void lift_kernel(const float* __restrict__ x,
                                                   float* __restrict__ t) {
    int idx = blockIdx.x * 256 + threadIdx.x;
    if (idx >= BATCH * SEQ * DMOD) return;
    int dg = idx & (DMOD - 1);
    int s  = (idx >> 9) % SEQ;
    int b  = idx / (DMOD * SEQ);
    int h  = dg >> 6;
    int d  = dg & (DK - 1);
    float val = x[idx];
    val = val > 0.0f ? val : 0.0f;
    t[(((size_t)(b * NH + h)) * SEQ + s) * DK + d] = log1pf(val);
}

// ---------------------------------------------------------------------------
// Kernel 2: q/k/v[row][o] = max_i ( t[row][i] + W[o][i] )   (max-plus matvec)
// One block of 256 threads handles 4 rows; W held transposed in LDS
// (Wt[i][o]) so the inner-loop reads are conflict-free.
// ---------------------------------------------------------------------------
__global__ __launch_bounds__(256) void qkv_kernel(const float* __restrict__ t,
                                                  const float* __restrict__ Wq,
                                                  const float* __restrict__ Wk,
                                                  const float* __restrict__ Wv,
                                                  float* __restrict__ q,
                                                  float* __restrict__ k,
                                                  float* __restrict__ v) {
    __shared__ float wqt[DK * DK];
    __shared__ float wkt[DK * DK];
    __shared__ float wvt[DK * DK];
    __shared__ float ts[4][DK];

    const int tid = threadIdx.x;
    for (int idx = tid; idx < DK * DK; idx += 256) {
        int o = idx / DK, i = idx % DK;
        wqt[i * DK + o] = Wq[idx];
        wkt[i * DK + o] = Wk[idx];
        wvt[i * DK + o] = Wv[idx];
    }
    const int row0 = blockIdx.x * 4;  // row in flattened [BH*SEQ]
    for (int idx = tid; idx < 4 * DK; idx += 256)
        ts[idx / DK][idx % DK] = t[(size_t)row0 * DK + idx];
    __syncthreads();

    const int r = tid >> 6;   // 0..3
    const int o = tid & 63;   // 0..63
    float aq = -__builtin_inff(), ak = -__builtin_inff(), av = -__builtin_inff();
#pragma unroll 4
    for (int i = 0; i < DK; ++i) {
        float ti = ts[r][i];
        aq = fmaxf(aq, ti + wqt[i * DK + o]);
        ak = fmaxf(ak, ti + wkt[i * DK + o]);
        av = fmaxf(av, ti + wvt[i * DK + o]);
    }
    size_t oi = (size_t)(row0 + r) * DK + o;
    q[oi] = aq;
    k[oi] = ak;
    v[oi] = av;
}

// ---------------------------------------------------------------------------
// Kernel 3 (dominant): fused tropical scores + max-plus context.
//   score[i][j] = -(max_d(q_i[d]-k_j[d]) - min_d(q_i[d]-k_j[d]))
//   ctx[i][d]   = max_j (score[i][j] + v[j][d])
// One block per (bh, 16-row i-tile); k/v 16-row tiles are pulled into LDS
// with CDNA5 async copies (GLOBAL_LOAD_ASYNC_TO_LDS_B128, ASYNCcnt), and
// all tile reads are ds_load_b128 (padded, conflict-free rows).
// ---------------------------------------------------------------------------
__global__ __launch_bounds__(256) void attn_kernel(const float* __restrict__ q,
                                                   const float* __restrict__ k,
                                                   const float* __restrict__ v,
                                                   float* __restrict__ attn,
                                                   float* __restrict__ ctxA) {
    __shared__ float qs[TI][DKP];
    __shared__ float ks[TJ][DKP];
    __shared__ float vs[TJ][DKP];
    __shared__ float ss[TI][TJ + 1];

    const int tid = threadIdx.x;
    const int bh  = blockIdx.x / (SEQ / TI);
    const int it  = blockIdx.x % (SEQ / TI);
    const int i0  = it * TI;

    // per-thread tile coordinates: one float4 of one row per thread
    const int lr = tid >> 4;         // row   0..15
    const int lc = (tid & 15) * 4;   // col   0..60 (16B aligned)

    // q tile load (vectorized b128)
    *(float4*)&qs[lr][lc] =
        *(const float4*)(q + ((size_t)bh * SEQ + i0 + lr) * DK + lc);

    // async source bases and LDS destinations (LDS addr = low 32 bits of
    // the generic shared-memory address, per the gfx1250 aperture rule)
    const float*   kbase = k + (size_t)bh * SEQ * DK + (size_t)lr * DK + lc;
    const float*   vbase = v + (size_t)bh * SEQ * DK + (size_t)lr * DK + lc;
    const uint32_t klds  = (uint32_t)(uintptr_t)&ks[lr][lc];
    const uint32_t vlds  = (uint32_t)(uintptr_t)&vs[lr][lc];

    // phase-1 mapping (one (i,j) score per thread) and
    // phase-2 mapping (4 contiguous d per thread of one i-row)
    const int si = tid >> 4;         // 0..15
    const int sj = tid & 15;         // 0..15
    const int ci = si;
    const int cd = sj * 4;

    float acc0 = -__builtin_inff(), acc1 = -__builtin_inff();
    float acc2 = -__builtin_inff(), acc3 = -__builtin_inff();

    for (int jt = 0; jt < SEQ / TJ; ++jt) {
        __syncthreads();  // everyone done reading ks/vs/ss of previous tile

        // CDNA5 async global -> LDS copy of the k/v tiles (16B per lane)
        {
            const uint64_t kaddr =
                (uint64_t)(uintptr_t)(kbase + (size_t)jt * TJ * DK);
            const uint64_t vaddr =
                (uint64_t)(uintptr_t)(vbase + (size_t)jt * TJ * DK);
            asm volatile("global_load_async_to_lds_b128 %0, %1, off"
                         :: "v"(klds), "v"(kaddr) : "memory");
            asm volatile("global_load_async_to_lds_b128 %0, %1, off"
                         :: "v"(vlds), "v"(vaddr) : "memory");
            asm volatile("s_wait_asynccnt 0" ::: "memory");
        }
        __syncthreads();  // all waves' tile data visible

        // phase 1: symmetric tropical distance for one (i,j) pair (b128 reads)
        float mx = -__builtin_inff(), mn = __builtin_inff();
        const float4* qrow = (const float4*)&qs[si][0];
        const float4* krow = (const float4*)&ks[sj][0];
#pragma unroll
        for (int d4 = 0; d4 < DK / 4; ++d4) {
            float4 qv = qrow[d4];
            float4 kv = krow[d4];
            float f0 = qv.x - kv.x, f1 = qv.y - kv.y;
            float f2 = qv.z - kv.z, f3 = qv.w - kv.w;
            mx = fmaxf(mx, fmaxf(fmaxf(f0, f1), fmaxf(f2, f3)));
            mn = fminf(mn, fminf(fminf(f0, f1), fminf(f2, f3)));
        }
        float sc = mn - mx;  // == -(mx - mn)
        ss[si][sj] = sc;
        attn[(size_t)bh * SEQ * SEQ + (size_t)(i0 + si) * SEQ + jt * TJ + sj] = sc;
        __syncthreads();

        // phase 2: max-plus combine with V tile (b128 reads)
#pragma unroll
        for (int j = 0; j < TJ; ++j) {
            float  s  = ss[ci][j];
            float4 vv = *(const float4*)&vs[j][cd];
            acc0 = fmaxf(acc0, s + vv.x);
            acc1 = fmaxf(acc1, s + vv.y);
            acc2 = fmaxf(acc2, s + vv.z);
            acc3 = fmaxf(acc3, s + vv.w);
        }
    }

    // expm1 + re-permute [BH,S,dk] -> [B*S, D] for the output GEMM
    const int b = bh >> 3, h = bh & 7;
    const int srow = i0 + ci;
    float4 r4 = make_float4(expm1f(acc0), expm1f(acc1), expm1f(acc2), expm1f(acc3));
    *(float4*)(ctxA + ((size_t)(b * SEQ + srow) * DMOD) + h * DK + cd) = r4;
}

// ---------------------------------------------------------------------------
// Kernel 4: out[m,n] = sum_k A[m,k] * Wout[n,k]   (A = expm1(context))
// Native-f32 WMMA: V_WMMA_F32_16X16X4_F32, one wave per 16x16 C tile.
// ---------------------------------------------------------------------------
__global__ __launch_bounds__(256) void gemm_kernel(const float* __restrict__ A,
                                                   const float* __restrict__ Wout,
                                                   float* __restrict__ out) {
    const int lane = threadIdx.x & 31;
    const int wave = threadIdx.x >> 5;
    const int tile = blockIdx.x * 8 + wave;
    const int tm = tile >> 5;          // 768/16 = 48 tiles in M
    const int tn = tile & 31;          // 512/16 = 32 tiles in N
    const int m0 = tm * 16, n0 = tn * 16;
    const bool hi = lane >= 16;
    const int  l  = lane & 15;
    const int  arow = m0 + l;
    const int  bcol = n0 + l;
    const int  koff = hi ? 2 : 0;

    const float* arp = A + (size_t)arow * DMOD;
    const float* brp = Wout + (size_t)bcol * DMOD;

    v8f c = {};
    for (int k0 = 0; k0 < DMOD; k0 += 4) {
        v2f a, b;
        a.x = arp[k0 + koff];
        a.y = arp[k0 + koff + 1];
        b.x = brp[k0 + koff];
        b.y = brp[k0 + koff + 1];
        c = __builtin_amdgcn_wmma_f32_16x16x4_f32(
                /*neg_a=*/false, a, /*neg_b=*/false, b,
                /*c_mod=*/(short)0, c, /*reuse_a=*/false, /*reuse_b=*/false);
    }

    const int orow = m0 + (hi ? 8 : 0);
    const int ocol = n0 + l;
#pragma unroll
    for (int r = 0; r < 8; ++r)
        out[(size_t)(orow + r) * DMOD + ocol] = c[r];
}

// ---------------------------------------------------------------------------
// Host-side launcher
// ---------------------------------------------------------------------------
extern "C" void kernel_launch(void* const* d_in, const int* in_sizes, int n_in,
                              void* d_out, int out_size, void* d_ws, size_t ws_size,
                              hipStream_t stream) {
    const float* x    = (const float*)d_in[0];
    const float* Wq   = (const float*)d_in[1];
    const float* Wk   = (const float*)d_in[2];
    const float* Wv   = (const float*)d_in[3];
    const float* Wout = (const float*)d_in[4];
    // d_in[5] = n_heads (constant 8, baked in)

    float* out  = (float*)d_out;                       // [B,S,D] = 393216
    float* attn = out + (size_t)BATCH * SEQ * DMOD;    // [BH,S,S] = 2359296

    const size_t N = (size_t)BH * SEQ * DK;            // 393216 per tensor
    float* t    = (float*)d_ws;
    float* q    = t + N;
    float* k    = q + N;
    float* v    = k + N;
    float* ctxA = v + N;                               // [B*S, D]

    lift_kernel<<<(BATCH * SEQ * DMOD + 255) / 256, 256, 0, stream>>>(x, t);
    qkv_kernel<<<BH * SEQ / 4, 256, 0, stream>>>(t, Wq, Wk, Wv, q, k, v);
    attn_kernel<<<BH * (SEQ / TI), 256, 0, stream>>>(q, k, v, attn, ctxA);
    gemm_kernel<<<(48 * 32) / 8, 256, 0, stream>>>(ctxA, Wout, out);
}